// WeightedRCQDecoder_13640816132469
// MI455X (gfx1250) — compile-verified
//
#include <hip/hip_runtime.h>
#include <stdint.h>

// LDPC WRCQ decoder for MI455X (gfx1250).
//
// Design (memory-bound, L2-resident):
//  - msg[] holds the edge messages in COLUMN-MAJOR variable layout:
//      edge e = 3*v + k  lives at msg[k*N + v]
//    so the variable pass is 4 coalesced streams in, 3 coalesced streams out.
//  - A check-CSR (6 edge ids per check, stored column-major chk_edges[j*M+m])
//    is built once with atomics. All check-side math (min1/min2/parity) is
//    order-independent; the min-tie-break uses the lowest ORIGINAL edge id,
//    exactly matching the reference, so atomic slot order is irrelevant ->
//    deterministic output.
//  - Check pass reads and rewrites each edge message in place (each edge
//    belongs to exactly one check) -> no separate c2v array.
//  - Working set (~85 MB) fits in the 192 MB L2; per-iteration traffic
//    (~130 MB) runs out of L2. No WMMA: there is no matmul here, ~2.5 GFLOP
//    total vs ~1.3 GB of L2 traffic -> pure bandwidth problem.
//  - CDNA5 path: variable pass stages its 4 streams into LDS via the Tensor
//    Data Mover (tensor_load_to_lds + s_wait_tensorcnt), issued by wave 0.

#define DV 3
#define DC 6
#define KQ 8
#define VP_BLOCK 256
#define VP_VPT 4
#define CHUNK (VP_BLOCK * VP_VPT)   // 1024 variables per workgroup

#if defined(__HIP_DEVICE_COMPILE__) && defined(__gfx1250__) && \
    __has_builtin(__builtin_amdgcn_tensor_load_to_lds)
#define USE_TDM 1
#else
#define USE_TDM 0
#endif

#if USE_TDM
typedef __attribute__((ext_vector_type(4))) unsigned int v4u_t;
typedef __attribute__((ext_vector_type(8))) int v8i_t;
typedef __attribute__((ext_vector_type(4))) int v4i_t;

__device__ __forceinline__ unsigned lds_off(const void* p) {
  // generic -> LDS addrspace cast, then take the 32-bit LDS byte offset
  return (unsigned)(unsigned long long)(__attribute__((address_space(3))) const void*)p;
}

// 1-D TDM load: copy `tile` f32 elements from gsrc into LDS at lds_byte.
// tensor_dim0 = avail (elements remaining in the stream) so the TDM's
// out-of-bounds handling zero-fills the tail tile.
__device__ __forceinline__ void tdm_load_1d(unsigned lds_byte, const float* gsrc,
                                            unsigned avail, unsigned tile) {
  unsigned long long ga = (unsigned long long)(uintptr_t)gsrc;
  v4u_t g0;
  g0.x = 1u;                                      // count=1 (valid), user mode
  g0.y = lds_byte;                                // lds_addr
  g0.z = (unsigned)ga;                            // global_addr[31:0]
  g0.w = (unsigned)((ga >> 32) & 0x01FFFFFFull)   // global_addr[56:32]
         | (2u << 30);                            // type=2 ("image")
  v8i_t g1;
  g1[0] = (int)(2u << 16);                        // data_size=2 -> 4 bytes
  g1[1] = (int)((avail & 0xFFFFu) << 16);         // tensor_dim0[15:0]
  g1[2] = (int)(((avail >> 16) & 0xFFFFu)         // tensor_dim0[31:16]
                | (1u << 16));                    // tensor_dim1 = 1
  g1[3] = (int)((tile & 0xFFFFu) << 16);          // tile_dim0
  g1[4] = 0;                                      // tile_dim1=0, tile_dim2=0
  g1[5] = (int)avail;                             // tensor_dim0_stride[31:0]
  g1[6] = 0;                                      // stride hi / dim1 stride lo
  g1[7] = 0;
  v4i_t z4 = {0, 0, 0, 0};
  v8i_t z8 = {0, 0, 0, 0, 0, 0, 0, 0};
  // 6-arg form (clang-23 / therock headers): groups 0..3, spare group, cpol
  __builtin_amdgcn_tensor_load_to_lds(g0, g1, z4, z4, z8, 0);
}
#endif

// ---- setup: v2c0 = llr[var] replicated into the 3 columns; zero CSR counters
__global__ void k_init(const float* __restrict__ llr, float* __restrict__ msg,
                       int* __restrict__ cnt, int N, int M) {
  int i = blockIdx.x * blockDim.x + threadIdx.x;
  if (i < N) {
    float L = llr[i];
    msg[i] = L;
    msg[(size_t)N + i] = L;
    msg[(size_t)2 * N + i] = L;
  }
  if (i < M) cnt[i] = 0;
}

// ---- build check CSR: chk_edges[slot*M + chk] = original edge id
__global__ void k_csr(const int* __restrict__ chk_idx, int* __restrict__ cnt,
                      int* __restrict__ chk_edges, int E, int M) {
  int e = blockIdx.x * blockDim.x + threadIdx.x;
  if (e >= E) return;
  int c = chk_idx[e];
  int pos = atomicAdd(&cnt[c], 1);
  chk_edges[(size_t)pos * M + c] = e;
}

// ---- check-node pass: min-sum with exclusion + RCQ quantize, in place
__global__ void k_check(const int* __restrict__ chk_edges, float* __restrict__ msg,
                        const float* __restrict__ beta_arr,
                        const float* __restrict__ thr_arr,
                        int t, int N, int M) {
  int m = blockIdx.x * blockDim.x + threadIdx.x;
  if (m >= M) return;
  const float b = beta_arr[t];                 // uniform -> s_load
  const float* thr = thr_arr + t * KQ;         // uniform -> s_load
  const float th0 = thr[0], th1 = thr[1], th2 = thr[2], th3 = thr[3];
  const float th4 = thr[4], th5 = thr[5], th6 = thr[6], th7 = thr[7];

  int e[DC];
  int adr[DC];
  float val[DC];
#pragma unroll
  for (int j = 0; j < DC; ++j) {
    int ej = chk_edges[(size_t)j * M + m];     // coalesced across checks
    e[j] = ej;
    int v = ej / 3;                            // var_idx structure: e = 3v+k
    int k = ej - 3 * v;
    adr[j] = k * N + v;                        // column-major msg address
    val[j] = msg[adr[j]];                      // L2-resident gather
  }

  float mag[DC];
  float m1 = 3.0e38f;
  unsigned par = 0;
#pragma unroll
  for (int j = 0; j < DC; ++j) {
    mag[j] = fabsf(val[j]);
    par ^= (val[j] < 0.0f) ? 1u : 0u;
    m1 = fminf(m1, mag[j]);
  }
  // first edge (lowest original id) attaining min1 — matches reference tie-break
  int efirst = 0x7fffffff;
#pragma unroll
  for (int j = 0; j < DC; ++j)
    efirst = min(efirst, (mag[j] == m1) ? e[j] : 0x7fffffff);
  // min over all edges except efirst
  float m2 = 3.0e38f;
#pragma unroll
  for (int j = 0; j < DC; ++j)
    m2 = fminf(m2, (e[j] == efirst) ? 3.0e38f : mag[j]);
  // DC==6 so m2 is always finite (degree-1 guard unnecessary)
  const float tot = (par & 1u) ? -1.0f : 1.0f;

#pragma unroll
  for (int j = 0; j < DC; ++j) {
    float sgn = (val[j] < 0.0f) ? -1.0f : 1.0f;
    float mv = (e[j] == efirst) ? m2 : m1;
    float raw = b * (tot * sgn) * mv;
    float aab = fabsf(raw);
    // searchsorted(thr, |raw|, right)-1 clipped == largest thr[q] <= |raw|
    float tq = th0;
    tq = (aab >= th1) ? th1 : tq;
    tq = (aab >= th2) ? th2 : tq;
    tq = (aab >= th3) ? th3 : tq;
    tq = (aab >= th4) ? th4 : tq;
    tq = (aab >= th5) ? th5 : tq;
    tq = (aab >= th6) ? th6 : tq;
    tq = (aab >= th7) ? th7 : tq;
    msg[adr[j]] = (raw < 0.0f) ? -tq : tq;     // c2v, in place
  }
}

// ---- variable-node pass: 4 coalesced streams in (via TDM->LDS), 3 out
__global__ void k_var(float* __restrict__ msg, const float* __restrict__ llr,
                      const float* __restrict__ alpha_arr, int t, int N,
                      float* __restrict__ out_dec, float* __restrict__ out_post,
                      float* __restrict__ out_T, int last, int Tval) {
  const int v0 = blockIdx.x * CHUNK;
  const float a = alpha_arr[t];
#if USE_TDM
  __shared__ float sm[4 * CHUNK];
  if ((threadIdx.x >> 5) == 0) {               // wave 0 issues the DMA
    unsigned base = lds_off(sm);
    unsigned avail = (unsigned)(N - v0);       // OOB tail -> TDM zero-fill
    tdm_load_1d(base + 0u * CHUNK * 4u, msg + v0, avail, CHUNK);
    tdm_load_1d(base + 1u * CHUNK * 4u, msg + (size_t)N + v0, avail, CHUNK);
    tdm_load_1d(base + 2u * CHUNK * 4u, msg + (size_t)2 * N + v0, avail, CHUNK);
    tdm_load_1d(base + 3u * CHUNK * 4u, llr + v0, avail, CHUNK);
    __builtin_amdgcn_s_wait_tensorcnt(0);      // LDS writes complete
  }
  __syncthreads();                             // visible to all waves
#endif
#pragma unroll
  for (int u = 0; u < VP_VPT; ++u) {
    int idx = (int)threadIdx.x + u * VP_BLOCK;
    int v = v0 + idx;
    if (v < N) {
#if USE_TDM
      float c0 = sm[idx];
      float c1 = sm[CHUNK + idx];
      float c2 = sm[2 * CHUNK + idx];
      float L  = sm[3 * CHUNK + idx];
#else
      float c0 = msg[v];
      float c1 = msg[(size_t)N + v];
      float c2 = msg[(size_t)2 * N + v];
      float L  = llr[v];
#endif
      float post = L + c0 + c1 + c2;
      msg[v]                 = a * (post - c0);
      msg[(size_t)N + v]     = a * (post - c1);
      msg[(size_t)2 * N + v] = a * (post - c2);
      if (last) {
        out_post[v] = post;
        out_dec[v] = (post < 0.0f) ? 1.0f : 0.0f;
        if (v == 0) out_T[0] = (float)Tval;
      }
    }
  }
}

extern "C" void kernel_launch(void* const* d_in, const int* in_sizes, int n_in,
                              void* d_out, int out_size, void* d_ws, size_t ws_size,
                              hipStream_t stream) {
  const float* llr     = (const float*)d_in[0];
  const int*   chk_idx = (const int*)d_in[2];   // var_idx (d_in[1]) is implicit: e=3v+k
  const float* beta    = (const float*)d_in[3];
  const float* alpha   = (const float*)d_in[4];
  const float* thr     = (const float*)d_in[5];
  const int N = in_sizes[0];
  const int E = in_sizes[2];
  const int T = in_sizes[3];
  const int M = E / DC;                         // regular (3,6) graph

  // workspace: msg (E f32) | chk_edges (E i32) | cnt (M i32)  ~= 54.5 MB
  float* msg       = (float*)d_ws;
  int*   chk_edges = (int*)((char*)d_ws + (size_t)E * 4);
  int*   cnt       = (int*)((char*)d_ws + (size_t)E * 8);

  // outputs: decoded (N) | posterior (N) | T (1), all as f32 per harness convention
  float* out_dec  = (float*)d_out;
  float* out_post = out_dec + N;
  float* out_T    = out_post + N;

  k_init<<<(N + 255) / 256, 256, 0, stream>>>(llr, msg, cnt, N, M);
  k_csr<<<(E + 255) / 256, 256, 0, stream>>>(chk_idx, cnt, chk_edges, E, M);
  for (int t = 0; t < T; ++t) {
    k_check<<<(M + 255) / 256, 256, 0, stream>>>(chk_edges, msg, beta, thr, t, N, M);
    k_var<<<(N + CHUNK - 1) / CHUNK, VP_BLOCK, 0, stream>>>(
        msg, llr, alpha, t, N, out_dec, out_post, out_T, (t == T - 1) ? 1 : 0, T);
  }
  (void)n_in; (void)out_size; (void)ws_size;
}